// MultistreamLoss_62191126446118
// MI455X (gfx1250) — compile-verified
//
#include <hip/hip_runtime.h>
#include <hip/hip_bf16.h>
#include <math.h>

// ---------------------------------------------------------------------------
// MultistreamLoss on gfx1250 (MI455X).
//
// ranking(p,l) = sum_{i<j} relu(margin - (p_j-p_i)*sign(l_j-l_i))
//   h is symmetric, h_ii = margin  =>  ranking = 0.5*(full_sum - n*margin)
// grad_k ranking(pred,l) = sum_m sign(l_m-l_k) * 1{margin - (p_m-p_k)*sign(l_m-l_k) > 0}
//   (full-matrix row reduction; diagonal term has sign 0 so no exclusion needed)
//
// Difference tiles (p_j - p_i) and (l_j - l_i) over a 16x16 (i,j) tile are the
// rank-2 product A(16x4) x B(4x16) with A = [1, p_i, 0, 0], B = [p_j; -1; 0; 0],
// computed with V_WMMA_F32_16X16X4_F32. D layout: lanes 0-15 hold column N=lane,
// VGPR v holds row M=v (lanes 16-31: M=8+v, N=lane-16) -> per-row gradient
// reduction is a shuffle-reduce within each 16-lane half.
// ---------------------------------------------------------------------------

typedef __attribute__((ext_vector_type(2))) float v2f;
typedef __attribute__((ext_vector_type(8))) float v8f;

#define N        4096
#define NBLK     256          // N / 16
#define NSTREAM  4
#define NTASK    (5 * NBLK)   // (4 score streams + pred) x row-blocks
#define MARGIN_F 2.0f
#define EPS_F    1e-4f

__global__ __launch_bounds__(256)
void ms_pairwise_kernel(const float* __restrict__ scores,  // [4, N]
                        const float* __restrict__ pred,    // [N]
                        const float* __restrict__ label,   // [N]
                        float* __restrict__ partials,      // [NTASK] hinge partial sums
                        float* __restrict__ grad)          // [N] d(ranking)/d(pred)
{
    const int lane  = threadIdx.x & 31;
    const int wave  = threadIdx.x >> 5;
    const int task  = blockIdx.x * (blockDim.x >> 5) + wave;   // wave-uniform
    const int strm  = task >> 8;        // 0..3 = scores rows, 4 = pred
    const int rb    = task & 255;       // row block
    const int ib    = rb << 4;          // i base

    const float* __restrict__ p = (strm < NSTREAM) ? (scores + strm * N) : pred;

    const int  li = lane & 15;
    const bool lo = (lane < 16);

    // A operand (16x4): lanes 0-15 -> {K0=1, K1=p_i[M]}, lanes 16-31 -> {K2=0, K3=0}
    v2f a_p, a_l;
    a_p.x = lo ? 1.0f : 0.0f;
    a_p.y = lo ? p[ib + li] : 0.0f;
    a_l.x = a_p.x;
    a_l.y = lo ? label[ib + li] : 0.0f;

    v8f   racc = {};     // per-row grad partials (row = v + 8*(lane/16))
    float hin  = 0.0f;   // hinge partial sum

    for (int jb = 0; jb < N; jb += 16) {
        const float pj = p[jb + li];
        const float lj = label[jb + li];
        // B operand (4x16): lanes 0-15 -> {K0=p_j[N], K1=-1}, lanes 16-31 -> 0
        v2f b_p, b_l;
        b_p.x = lo ? pj    : 0.0f;
        b_p.y = lo ? -1.0f : 0.0f;
        b_l.x = lo ? lj    : 0.0f;
        b_l.y = b_p.y;

        v8f c0 = {};
        // D[M][N] = p[jb+N] - p[ib+M]
        v8f pd = __builtin_amdgcn_wmma_f32_16x16x4_f32(
            false, a_p, false, b_p, (short)0, c0, false, false);
        // D[M][N] = l[jb+N] - l[ib+M]
        v8f ld = __builtin_amdgcn_wmma_f32_16x16x4_f32(
            false, a_l, false, b_l, (short)0, c0, false, false);

#pragma unroll
        for (int v = 0; v < 8; ++v) {
            const float sl = ld[v];
            const float s  = (sl > 0.0f) ? 1.0f : ((sl < 0.0f) ? -1.0f : 0.0f);
            const float t  = fmaf(-pd[v], s, MARGIN_F);   // margin - pd*s
            hin += fmaxf(t, 0.0f);
            racc[v] += (t > 0.0f) ? s : 0.0f;             // relu'(0)=0 per JAX
        }
    }

    // Row-gradient: sum columns => reduce across each 16-lane half.
#pragma unroll
    for (int m = 8; m >= 1; m >>= 1) {
#pragma unroll
        for (int v = 0; v < 8; ++v)
            racc[v] += __shfl_xor(racc[v], m, 32);
    }
    if (strm == NSTREAM && (lane & 15) == 0) {
        const int rowoff = ib + ((lane >> 4) << 3);  // lane0 -> rows 0..7, lane16 -> 8..15
#pragma unroll
        for (int v = 0; v < 8; ++v)
            grad[rowoff + v] = racc[v];
    }

    // Hinge partial: reduce across the whole wave.
#pragma unroll
    for (int m = 16; m >= 1; m >>= 1)
        hin += __shfl_xor(hin, m, 32);
    if (lane == 0)
        partials[task] = hin;
}

__global__ __launch_bounds__(256)
void ms_finalize_kernel(const float* __restrict__ pred,
                        const float* __restrict__ label,
                        const float* __restrict__ partials,  // [NTASK]
                        const float* __restrict__ grad,      // [N]
                        float* __restrict__ out)
{
    __shared__ float sm[256];
    const int t = threadIdx.x;

    float vals[7];
    // [0] S2 = sum (pred-label)^2 ; [1] G2 = sum grad^2 ; [2..6] per-stream hinge sums
    float s2 = 0.0f, g2 = 0.0f;
    for (int i = t; i < N; i += 256) {
        const float d = pred[i] - label[i];
        s2 += d * d;
        const float g = grad[i];
        g2 += g * g;
    }
    vals[0] = s2;
    vals[1] = g2;
#pragma unroll
    for (int s = 0; s < 5; ++s)
        vals[2 + s] = partials[s * NBLK + t];   // NBLK == blockDim

    float res[7];
#pragma unroll
    for (int k = 0; k < 7; ++k) {
        sm[t] = vals[k];
        __syncthreads();
        for (int off = 128; off >= 1; off >>= 1) {
            if (t < off) sm[t] += sm[t + off];
            __syncthreads();
        }
        res[k] = sm[0];
        __syncthreads();
    }

    if (t == 0) {
        const float S2 = res[0];
        const float G2 = res[1];
        float rank[5];
#pragma unroll
        for (int s = 0; s < 5; ++s)
            rank[s] = 0.5f * (res[2 + s] - (float)N * MARGIN_F);  // triangular sum
        const float branch_loss = rank[0] + rank[1] + rank[2] + rank[3]; // weights = 1
        const float mse    = S2 / (float)N;
        const float g1     = (2.0f / (float)N) * sqrtf(S2);  // ||2(pred-label)/n||
        const float g2n    = sqrtf(G2);                      // ||d ranking/d pred||
        const float beta_w = g1 / (g2n + EPS_F);
        // ALPHA = 1/4, BETA = 1
        out[0] = 0.25f * branch_loss + (mse + beta_w * rank[4]);
    }
}

extern "C" void kernel_launch(void* const* d_in, const int* in_sizes, int n_in,
                              void* d_out, int out_size, void* d_ws, size_t ws_size,
                              hipStream_t stream) {
    const float* scores = (const float*)d_in[0];   // [4, 4096]
    const float* pred   = (const float*)d_in[1];   // [4096]
    const float* label  = (const float*)d_in[2];   // [4096]
    float* partials = (float*)d_ws;                // NTASK floats
    float* grad     = partials + NTASK;            // N floats
    float* out      = (float*)d_out;

    // 1280 wave-tasks, 8 waves (256 threads) per block -> 160 blocks exactly.
    ms_pairwise_kernel<<<dim3(NTASK / 8), dim3(256), 0, stream>>>(
        scores, pred, label, partials, grad);
    ms_finalize_kernel<<<dim3(1), dim3(256), 0, stream>>>(
        pred, label, partials, grad, out);
}